// GraphSAGE_83408264888606
// MI455X (gfx1250) — compile-verified
//
#include <hip/hip_runtime.h>

typedef __attribute__((ext_vector_type(2))) float v2f;
typedef __attribute__((ext_vector_type(8))) float v8f;

#define NN 100000
#define DD 128

// ---------------- zero ----------------
__global__ void k_zero(float* __restrict__ p, long n) {
  long i = (long)blockIdx.x * blockDim.x + threadIdx.x;
  long stride = (long)gridDim.x * blockDim.x;
  for (; i < n; i += stride) p[i] = 0.f;
}

// ---------------- out-degree (as float, via HW f32 atomics) ----------------
__global__ void k_deg(const int* __restrict__ src, float* __restrict__ deg, int E) {
  int e = blockIdx.x * blockDim.x + threadIdx.x;
  if (e < E) unsafeAtomicAdd(&deg[src[e]], 1.0f);
}

__global__ void k_inv(float* __restrict__ deg, int n) {
  int i = blockIdx.x * blockDim.x + threadIdx.x;
  if (i < n) deg[i] = 1.0f / fmaxf(deg[i], 1.0f);
}

// ---------------- edge scatter: agg[dst] += feat[src] * inv[src] ----------------
// one wave32 per edge; lane handles 4 contiguous features (float4 coalesced load)
__global__ void __launch_bounds__(256) k_scatter(
    const float* __restrict__ feat, const int* __restrict__ src,
    const int* __restrict__ dst, const float* __restrict__ inv,
    float* __restrict__ agg, int E) {
  int wave = threadIdx.x >> 5, lane = threadIdx.x & 31;
  int e = blockIdx.x * 8 + wave;
  if (e >= E) return;
  int s = src[e], d = dst[e];
  float w = inv[s];
  const float4 v = *(const float4*)(feat + (long)s * DD + lane * 4);
  float* o = agg + (long)d * DD + lane * 4;
  unsafeAtomicAdd(o + 0, v.x * w);
  unsafeAtomicAdd(o + 1, v.y * w);
  unsafeAtomicAdd(o + 2, v.z * w);
  unsafeAtomicAdd(o + 3, v.w * w);
}

// ---------------- WMMA GEMM: out = [fx | fa] @ W.T + bias, optional relu ----------
// blockIdx.y selects 64-output half; that W half (64x256 f32 = 64KB) is staged in
// LDS PRE-SWIZZLED into B-fragment lane order so the inner loop is a conflict-free
// lane-consecutive ds_load_b64 per (k-step, j-tile). Rolled k-loop to avoid the
// register-pressure explosion (spills) that full unrolling caused.
__global__ void __launch_bounds__(256) k_gemm(
    const float* __restrict__ fx, const float* __restrict__ fa,
    const float* __restrict__ W, const float* __restrict__ bias,
    float* __restrict__ out, int relu) {
  __shared__ float wlds[64 * 256];  // 65536 bytes

  const int jbase = blockIdx.y * 64;
  const int t = threadIdx.x;

  // Fill LDS in B-fragment order: e = (((jt*64 + k0)*32 + lane)*2 + v)
  // fragment element (lane, v) = W[jbase + jt*16 + (lane&15)][k0*4 + v + 2*(lane>>4)]
#pragma unroll 1
  for (int it = 0; it < 64; ++it) {
    int e = t + it * 256;
    int v = e & 1;
    int ln = (e >> 1) & 31;
    int k0 = (e >> 6) & 63;
    int jt = e >> 12;
    int j = jbase + jt * 16 + (ln & 15);
    int k = k0 * 4 + v + 2 * (ln >> 4);
    wlds[e] = W[j * 256 + k];
  }
  __syncthreads();

  const int wave = t >> 5, lane = t & 31;
  const int m = lane & 15, hi = lane >> 4;
  const int ntiles = NN / 16;  // 6250

#pragma unroll 1
  for (int tile = blockIdx.x * 8 + wave; tile < ntiles; tile += gridDim.x * 8) {
    const int row = tile * 16 + m;

    v8f acc[4];
#pragma unroll
    for (int jt = 0; jt < 4; ++jt) {
      float bj = bias[jbase + jt * 16 + m];
#pragma unroll
      for (int r = 0; r < 8; ++r) acc[jt][r] = bj;
    }

    // k in [0,128): self features fx
    const float* pa = fx + (long)row * DD + 2 * hi;
#pragma unroll 1
    for (int k0 = 0; k0 < 32; ++k0) {
      v2f a = *(const v2f*)(pa + k0 * 4);
#pragma unroll
      for (int jt = 0; jt < 4; ++jt) {
        v2f b = *(const v2f*)(&wlds[((jt * 64 + k0) * 32 + lane) * 2]);
        acc[jt] = __builtin_amdgcn_wmma_f32_16x16x4_f32(
            false, a, false, b, (short)0, acc[jt], false, false);
      }
    }
    // k in [128,256): aggregated features fa
    const float* pb = fa + (long)row * DD + 2 * hi;
#pragma unroll 1
    for (int k0 = 32; k0 < 64; ++k0) {
      v2f a = *(const v2f*)(pb + (k0 - 32) * 4);
#pragma unroll
      for (int jt = 0; jt < 4; ++jt) {
        v2f b = *(const v2f*)(&wlds[((jt * 64 + k0) * 32 + lane) * 2]);
        acc[jt] = __builtin_amdgcn_wmma_f32_16x16x4_f32(
            false, a, false, b, (short)0, acc[jt], false, false);
      }
    }

    // epilogue: C/D layout -> out[node, j]
#pragma unroll
    for (int jt = 0; jt < 4; ++jt) {
      int j = jbase + jt * 16 + m;
#pragma unroll
      for (int r = 0; r < 8; ++r) {
        float val = acc[jt][r];
        if (relu) val = fmaxf(val, 0.f);
        out[(long)(tile * 16 + r + 8 * hi) * DD + j] = val;
      }
    }
  }
}

extern "C" void kernel_launch(void* const* d_in, const int* in_sizes, int n_in,
                              void* d_out, int out_size, void* d_ws, size_t ws_size,
                              hipStream_t stream) {
  const float* x  = (const float*)d_in[0];
  const int* src  = (const int*)d_in[1];
  const int* dst  = (const int*)d_in[2];
  const float* W0 = (const float*)d_in[3];
  const float* b0 = (const float*)d_in[4];
  const float* W1 = (const float*)d_in[5];
  const float* b1 = (const float*)d_in[6];
  float* out = (float*)d_out;
  const int E = in_sizes[1];

  float* deg = (float*)d_ws;                 // N floats (becomes inv-degree)
  float* agg = deg + NN;                     // N*128 floats
  float* h   = agg + (size_t)NN * DD;        // N*128 floats

  // zero deg + agg (contiguous)
  k_zero<<<2048, 256, 0, stream>>>(deg, (long)NN * (DD + 1));
  k_deg<<<(E + 255) / 256, 256, 0, stream>>>(src, deg, E);
  k_inv<<<(NN + 255) / 256, 256, 0, stream>>>(deg, NN);

  // layer 0
  k_scatter<<<(E + 7) / 8, 256, 0, stream>>>(x, src, dst, deg, agg, E);
  dim3 g(256, 2);
  k_gemm<<<g, 256, 0, stream>>>(x, agg, W0, b0, h, 1);

  // layer 1
  k_zero<<<2048, 256, 0, stream>>>(agg, (long)NN * DD);
  k_scatter<<<(E + 7) / 8, 256, 0, stream>>>(h, src, dst, deg, agg, E);
  k_gemm<<<g, 256, 0, stream>>>(h, agg, W1, b1, out, 0);
}